// JointDecoder_57698590654834
// MI455X (gfx1250) — compile-verified
//
#include <hip/hip_runtime.h>
#include <math.h>
#include <stdint.h>

// ---- problem constants (from reference) ----
#define DBINS     128
#define BEV       128
#define CELL      (BEV * BEV)        // 16384
#define TABLE_N   (3 * CELL)         // 49152 floats = 192 KB (fits 320KB WGP LDS)
#define RES_F     0.025f
#define MIN_DEP   0.1125f            // Z_MIN + RES/2
#define LOG_U3    (-1.0986f)
#define EPS_F     1e-7f

typedef unsigned int __attribute__((ext_vector_type(4))) v4u;
typedef int          __attribute__((ext_vector_type(4))) v4i;
typedef int          __attribute__((ext_vector_type(8))) v8i;

// =====================================================================
// Kernel 1: bev_logits (B,2,128,128) -> log class-prob LUT (B,3,128,128)
// Precomputing the logs removes ~150M v_log ops from the hot loop.
// =====================================================================
__global__ void build_logtab_kernel(const float* __restrict__ bev,
                                    float* __restrict__ tab, int total) {
  int i = blockIdx.x * blockDim.x + threadIdx.x;
  if (i >= total) return;
  int b  = i / CELL;
  int zx = i - b * CELL;
  float l0 = bev[(size_t)(b * 2 + 0) * CELL + zx];
  float l1 = bev[(size_t)(b * 2 + 1) * CELL + zx];
  float p0 = 1.0f / (1.0f + __expf(-l0));
  float p1 = 1.0f / (1.0f + __expf(-l1));
  p0 = fminf(fmaxf(p0, EPS_F), 1.0f - EPS_F);
  p1 = fminf(fmaxf(p1, EPS_F), 1.0f - EPS_F);
  float* t = tab + (size_t)b * TABLE_N + zx;
  t[0]        = __logf(1.0f - p1);         // class 0: 1 - p1
  t[CELL]     = __logf(p1 * p0);           // class 1: p1*p0
  t[2 * CELL] = __logf(p1 * (1.0f - p0));  // class 2: p1*(1-p0)
}

// =====================================================================
// Kernel 2: main fused decoder.
//  - TDM DMA stages the 192KB per-batch LUT into LDS (one DMA / block)
//  - single global read of depth logits into a 128-VGPR register array
//    (192KB LDS already caps us at 1 WG / WGP -> registers are free)
//  - register-resident log-softmax, voxelize + LDS gather, b128 stores
// =====================================================================
__global__ void __launch_bounds__(512)
joint_decoder_kernel(const float* __restrict__ depth_logits,
                     const float* __restrict__ logtab,
                     const float* __restrict__ invK,
                     float* __restrict__ out, int H, int W) {
  extern __shared__ float tab[];  // TABLE_N floats = 192 KB dynamic LDS
  const int b  = blockIdx.y;
  const int HW = H * W;
  const float* __restrict__ gtab = logtab + (size_t)b * TABLE_N;

  // ---------------- stage LUT into LDS via Tensor Data Mover ----------------
#if __has_builtin(__builtin_amdgcn_tensor_load_to_lds)
  if (threadIdx.x < 32) {  // wave 0 only: TDM issues once per wave (EXEC ignored)
    unsigned long long ga  = (unsigned long long)(uintptr_t)gtab;
    unsigned int       lds = (unsigned int)(unsigned long long)(uintptr_t)&tab[0];
    // D# group 0: count=1 | lds_addr | global_addr[56:0] | type=2
    v4u g0 = { 1u, lds, (unsigned)(ga & 0xffffffffu),
               (unsigned)(((ga >> 32) & 0x01ffffffu) | 0x80000000u) };
    // D# group 1: data_size=4B; tensor_dim0=tile_dim0=stride0=49152 (1-D tile)
    v8i g1 = { (int)0x00020000u,          // mask=0, data_size=2 (4B)
               (int)0xC0000000u,          // tensor_dim0[15:0]=49152 << 16
               (int)0x00010000u,          // tensor_dim0 hi=0, tensor_dim1=1
               (int)0xC0000000u,          // tensor_dim1 hi=0, tile_dim0=49152
               (int)1,                    // tile_dim1=1, tile_dim2=0
               (int)TABLE_N,              // tensor_dim0_stride lo
               0, 0 };
    v4i gz = { 0, 0, 0, 0 };
#if __has_include(<hip/amd_detail/amd_gfx1250_TDM.h>)
    v8i gz8 = { 0, 0, 0, 0, 0, 0, 0, 0 };
    __builtin_amdgcn_tensor_load_to_lds(g0, g1, gz, gz, gz8, 0);   // clang-23 form
#else
    __builtin_amdgcn_tensor_load_to_lds(g0, g1, gz, gz, 0);        // ROCm 7.2 form
#endif
    __builtin_amdgcn_s_wait_tensorcnt(0);
  }
#else
  for (int i = threadIdx.x; i < TABLE_N; i += 512) tab[i] = gtab[i];
#endif
  __syncthreads();

  // ---------------- per-pixel work ----------------
  const int pix = blockIdx.x * 512 + threadIdx.x;
  if (pix >= HW) return;
  const int h = pix / W;
  const int w = pix - h * W;
  const float fx = (float)w, fy = (float)h;

  const float* __restrict__ ik = invK + b * 16;  // (4,4) row-major, uniform per block
  const float cx = ik[0] * fx + ik[1] * fy + ik[2];
  const float cy = ik[4] * fx + ik[5] * fy + ik[6];
  const float cz = ik[8] * fx + ik[9] * fy + ik[10];

  const float* __restrict__ dl = depth_logits + (size_t)b * DBINS * HW + pix;

  // single coalesced read of all 128 depth logits into registers
  float lg[DBINS];
#pragma unroll
  for (int d = 0; d < DBINS; ++d) lg[d] = dl[(size_t)d * HW];

  // register-resident log-softmax
  float m = -INFINITY;
#pragma unroll
  for (int d = 0; d < DBINS; ++d) m = fmaxf(m, lg[d]);
  float s = 0.0f;
#pragma unroll
  for (int d = 0; d < DBINS; ++d) s += __expf(lg[d] - m);
  const float lse = m + __logf(s);

  // voxelize, branchless LDS gather, fused add, packed b128 stores
  float* __restrict__ op = out + (size_t)((size_t)b * HW + pix) * (DBINS * 3);
#pragma unroll 8
  for (int k = 0; k < DBINS / 4; ++k) {
    float buf[12];
#pragma unroll
    for (int j = 0; j < 4; ++j) {
      const int d = 4 * k + j;
      const float depth = MIN_DEP + (float)d * RES_F;
      const float px =  depth * cx;
      const float py = -depth * cy;
      const float pz = -depth * cz;
      const int v0 = (int)((px + 1.6f) * 40.0f);   // (x - X_MIN)/RES, trunc
      const int v1 = (int)( py         * 40.0f);
      const int v2 = (int)((pz + 3.3f) * 40.0f);   // (z + Z_MAX)/RES, trunc
      const bool valid = (v0 >= 0) && (v0 < BEV) && (v2 >= 0) && (v2 < BEV) && (v1 <= 0);
      const int base = valid ? (v2 * BEV + v0) : 0;  // safe_base, branchless
      const float t0 = tab[base];
      const float t1 = tab[CELL + base];
      const float t2 = tab[2 * CELL + base];
      const float dlp = lg[d] - lse;
      buf[3 * j + 0] = (valid ? t0 : LOG_U3) + dlp;
      buf[3 * j + 1] = (valid ? t1 : LOG_U3) + dlp;
      buf[3 * j + 2] = (valid ? t2 : LOG_U3) + dlp;
    }
    // 48 contiguous bytes per thread per chunk, 16B aligned -> b128 stores
    float4* o4 = (float4*)(op + (size_t)k * 12);
    o4[0] = make_float4(buf[0], buf[1], buf[2],  buf[3]);
    o4[1] = make_float4(buf[4], buf[5], buf[6],  buf[7]);
    o4[2] = make_float4(buf[8], buf[9], buf[10], buf[11]);
  }
}

// =====================================================================
extern "C" void kernel_launch(void* const* d_in, const int* in_sizes, int n_in,
                              void* d_out, int out_size, void* d_ws, size_t ws_size,
                              hipStream_t stream) {
  const float* depth_logits = (const float*)d_in[0];  // (B,128,H,W)
  const float* bev_logits   = (const float*)d_in[1];  // (B,2,128,128)
  const float* invK         = (const float*)d_in[2];  // (B,4,4)
  float*       out          = (float*)d_out;          // (B,H,W,128,3)

  const int B  = in_sizes[1] / (2 * CELL);
  const int HW = in_sizes[0] / (B * DBINS);
  int W = 1;
  while ((long long)W * W < (long long)HW) ++W;       // square images (setup: 256x256)
  const int H = HW / W;

  float* tab = (float*)d_ws;                          // B * 192 KB scratch LUT

  // allow 192KB dynamic LDS (host API call, capture-safe)
  (void)hipFuncSetAttribute((const void*)joint_decoder_kernel,
                            hipFuncAttributeMaxDynamicSharedMemorySize,
                            TABLE_N * (int)sizeof(float));

  const int total1 = B * CELL;
  build_logtab_kernel<<<(total1 + 255) / 256, 256, 0, stream>>>(bev_logits, tab, total1);

  dim3 grid((HW + 511) / 512, B);
  joint_decoder_kernel<<<grid, 512, TABLE_N * sizeof(float), stream>>>(
      depth_logits, tab, invK, out, H, W);
}